// GCN_Block__47296179863966
// MI455X (gfx1250) — compile-verified
//
#include <hip/hip_runtime.h>

// Problem constants (fixed by the reference).
#define BN    100000
#define EDGES 1600000
#define C     64        // C1 == C2 == 64

typedef float v2f __attribute__((ext_vector_type(2)));
typedef float v8f __attribute__((ext_vector_type(8)));

// ---------------------------------------------------------------------------
// Kernel 1: zero the output accumulator and the degree buffer (d_out / d_ws
// are poisoned or hold stale data from the previous replay).
// ---------------------------------------------------------------------------
__global__ void gcn_zero_kernel(float* __restrict__ y, float* __restrict__ deg) {
  int i = blockIdx.x * blockDim.x + threadIdx.x;
  if (i < BN * C) y[i] = 0.0f;
  if (i < BN)     deg[i] = 0.0f;
}

// ---------------------------------------------------------------------------
// Kernel 2: degree histogram over row endpoints.
// deg values are small integers; fp32 atomic add (global_atomic_add_f32) is
// exact here and avoids a separate int buffer + convert pass.
// ---------------------------------------------------------------------------
__global__ void gcn_deg_kernel(const long long* __restrict__ e,
                               float* __restrict__ deg) {
  int i = blockIdx.x * blockDim.x + threadIdx.x;
  if (i < EDGES) {
    int r = (int)e[i];              // row endpoint
    unsafeAtomicAdd(&deg[r], 1.0f); // -> global_atomic_add_f32
  }
}

// ---------------------------------------------------------------------------
// Kernel 3: dis[i] = deg[i] > 0 ? deg[i]^{-1/2} : 0
// ---------------------------------------------------------------------------
__global__ void gcn_dis_kernel(const float* __restrict__ deg,
                               float* __restrict__ dis) {
  int i = blockIdx.x * blockDim.x + threadIdx.x;
  if (i < BN) {
    float d = deg[i];
    dis[i] = (d > 0.0f) ? rsqrtf(d) : 0.0f;
  }
}

// ---------------------------------------------------------------------------
// Kernel 4: h = x @ W + b via V_WMMA_F32_16X16X4_F32 (fp32 exact path).
// One wave -> one 16-row tile of h (16x64), 4 x v8f accumulators,
// K stepped by 4 => 16 * 4 = 64 WMMAs per wave. BN % 16 == 0.
//
// A (16x4, MxK): lane L holds M = L&15; VGPR v holds K = 2*(L>=16) + v
//   -> per lane a contiguous float2 of x[row, kb+koff .. +1].
// B (4x16, KxN): VGPR v holds K = v + 2*(L>=16), N = L&15 (row striped
//   across lanes within one VGPR, halves split like the C/D layout).
// C/D (16x16): VGPR r holds M = r + 8*(L>=16), N = L&15.
// ---------------------------------------------------------------------------
__global__ void __launch_bounds__(128)
gcn_gemm_kernel(const float* __restrict__ x,
                const float* __restrict__ W,   // (C1, C2) row-major = (K, N)
                const float* __restrict__ b,
                float* __restrict__ h) {
  const int wave = threadIdx.x >> 5;
  const int lane = threadIdx.x & 31;
  const int tile = blockIdx.x * 4 + wave;     // 16-row tile index
  if (tile * 16 >= BN) return;                // wave-uniform guard

  const int m15  = lane & 15;
  const int hi   = lane >> 4;                 // 0: lanes 0-15, 1: lanes 16-31
  const int koff = hi * 2;

  // Accumulators, initialized with the bias (bias is constant per column N).
  v8f acc[4];
#pragma unroll
  for (int nt = 0; nt < 4; ++nt) {
    float bias = b[nt * 16 + m15];
#pragma unroll
    for (int r = 0; r < 8; ++r) acc[nt][r] = bias;
  }

  const float* xrow = x + (size_t)(tile * 16 + m15) * C;

#pragma unroll
  for (int kb = 0; kb < C; kb += 4) {
    v2f a;
    a.x = xrow[kb + koff];
    a.y = xrow[kb + koff + 1];
#pragma unroll
    for (int nt = 0; nt < 4; ++nt) {
      const int n = nt * 16 + m15;
      v2f bb;
      bb.x = W[(kb + koff)     * C + n];
      bb.y = W[(kb + koff + 1) * C + n];
      // 8 args: (neg_a, A, neg_b, B, c_mod, C, reuse_a, reuse_b)
      acc[nt] = __builtin_amdgcn_wmma_f32_16x16x4_f32(
          false, a, false, bb, (short)0, acc[nt], false, false);
    }
  }

  // Store D: VGPR r -> row m = r + 8*hi, column n = nt*16 + m15.
#pragma unroll
  for (int nt = 0; nt < 4; ++nt) {
    const int n = nt * 16 + m15;
#pragma unroll
    for (int r = 0; r < 8; ++r) {
      const int m = r + hi * 8;
      h[(size_t)(tile * 16 + m) * C + n] = acc[nt][r];
    }
  }
}

// ---------------------------------------------------------------------------
// Kernel 5: per-edge message + scatter-add.
// 16 lanes per edge, float4 per lane (coalesced 16B gather of h[row]),
// 4 hardware fp32 atomics into y[col]. h/y live in L2 (52 MB << 192 MB).
// ---------------------------------------------------------------------------
__global__ void gcn_edge_kernel(const long long* __restrict__ e,
                                const float* __restrict__ dis,
                                const float* __restrict__ h,
                                float* __restrict__ y) {
  long long gid  = (long long)blockIdx.x * blockDim.x + threadIdx.x;
  long long eidx = gid >> 4;
  if (eidx >= EDGES) return;
  const int c   = ((int)gid & 15) * 4;
  const int row = (int)e[eidx];
  const int col = (int)e[EDGES + eidx];
  const float norm = dis[row] * dis[col];

  const float4 hv = *reinterpret_cast<const float4*>(h + (size_t)row * C + c);
  float* yp = y + (size_t)col * C + c;
  unsafeAtomicAdd(yp + 0, norm * hv.x);
  unsafeAtomicAdd(yp + 1, norm * hv.y);
  unsafeAtomicAdd(yp + 2, norm * hv.z);
  unsafeAtomicAdd(yp + 3, norm * hv.w);
}

// ---------------------------------------------------------------------------
// Launch wrapper. Workspace layout (floats):
//   [0, BN)              deg
//   [BN, 2*BN)           dis
//   [2*BN, 2*BN+BN*C)    h
// Total ~26.4 MB.
// ---------------------------------------------------------------------------
extern "C" void kernel_launch(void* const* d_in, const int* in_sizes, int n_in,
                              void* d_out, int out_size, void* d_ws, size_t ws_size,
                              hipStream_t stream) {
  const float*     x  = (const float*)d_in[0];      // (BN, C1) fp32
  const long long* e  = (const long long*)d_in[1];  // (2, E)   int64
  const float*     W  = (const float*)d_in[2];      // (C1, C2) fp32
  const float*     bb = (const float*)d_in[3];      // (C2,)    fp32
  float* y = (float*)d_out;                         // (BN, C2) fp32

  float* deg = (float*)d_ws;
  float* dis = deg + BN;
  float* h   = dis + BN;

  (void)in_sizes; (void)n_in; (void)out_size; (void)ws_size;

  gcn_zero_kernel<<<(BN * C + 255) / 256, 256, 0, stream>>>(y, deg);
  gcn_deg_kernel<<<(EDGES + 255) / 256, 256, 0, stream>>>(e, deg);
  gcn_dis_kernel<<<(BN + 255) / 256, 256, 0, stream>>>(deg, dis);

  const int tiles = BN / 16;                        // 6250, exact
  gcn_gemm_kernel<<<(tiles + 3) / 4, 128, 0, stream>>>(x, W, bb, h);

  const long long ethreads = (long long)EDGES * 16; // 25.6M
  gcn_edge_kernel<<<(int)((ethreads + 255) / 256), 256, 0, stream>>>(e, dis, h, y);
}